// ImageCaptionGenerator_76321568850067
// MI455X (gfx1250) — compile-verified
//
#include <hip/hip_runtime.h>
#include <math.h>
#include <stdint.h>

#define B_   128
#define L_   64
#define C_   100
#define T_   32
#define E_   256
#define R_   512
#define A_   512
#define P_   128
#define VOC_ 20000
#define CE_  (C_ + E_)      // 356
#define LN_EPS 1e-3f

typedef __attribute__((ext_vector_type(2))) float        v2f;
typedef __attribute__((ext_vector_type(8))) float        v8f;
typedef __attribute__((ext_vector_type(4))) unsigned int v4u;
typedef __attribute__((ext_vector_type(8))) int          v8i;
typedef __attribute__((ext_vector_type(4))) int          v4i;

// ---------------------------------------------------------------------------
// fp32 WMMA fragment layouts (ISA 7.12.2, V_WMMA_F32_16X16X4_F32):
//  A 16x4 : lanes 0-15 -> K={k0,k0+1}, lanes 16-31 -> K={k0+2,k0+3}
//  B 4x16 : VGPR j holds rows {k0+2j (lanes0-15), k0+2j+1 (lanes16-31)}
//  C/D    : VGPR i -> rows {i (lanes0-15), i+8 (lanes16-31)}, col = lane&15
// One wave owns a full M=128 column strip (8 accumulator tiles) so each B
// fragment is reused by 8 WMMAs.
// ---------------------------------------------------------------------------
__device__ __forceinline__ v8f wmma4(v2f a, v2f b, v8f c) {
  return __builtin_amdgcn_wmma_f32_16x16x4_f32(false, a, false, b,
                                               (short)0, c, false, false);
}

// Generic strip GEMM: C[M,N] = A[M,K]*B[K,N] + bias (opt relu), lda=K, ldb=ldc=N.
// M = 128 * (#strips). One wave per (strip, 16-col tile).
template<int K, int N, int RELU>
__global__ void strip_gemm_kernel(const float* __restrict__ A,
                                  const float* __restrict__ B,
                                  const float* __restrict__ bias,
                                  float* __restrict__ C, int numWaves) {
  const int wid = blockIdx.x * (blockDim.x >> 5) + (threadIdx.x >> 5);
  if (wid >= numWaves) return;                 // wave-uniform, EXEC stays full
  constexpr int NT = N / 16;
  const int sid = wid / NT, nt = wid % NT;
  const int row0 = sid * 128, col0 = nt * 16;
  const int lane = threadIdx.x & 31;
  const int m = lane & 15, hi = lane >> 4;

  v8f acc[8];
#pragma unroll
  for (int i = 0; i < 8; ++i) acc[i] = (v8f){};

  for (int k0 = 0; k0 < K; k0 += 4) {
    v2f bf;
    bf.x = B[(size_t)(k0 + hi) * N + col0 + m];
    bf.y = B[(size_t)(k0 + 2 + hi) * N + col0 + m];
#pragma unroll
    for (int i = 0; i < 8; ++i) {
      v2f af = *(const v2f*)(A + (size_t)(row0 + i * 16 + m) * K + k0 + 2 * hi);
      acc[i] = wmma4(af, bf, acc[i]);
    }
  }
  const float bv = bias[col0 + m];
#pragma unroll
  for (int i = 0; i < 8; ++i)
#pragma unroll
    for (int r = 0; r < 8; ++r) {
      float v = acc[i][r] + bv;
      if (RELU) v = fmaxf(v, 0.0f);
      C[(size_t)(row0 + i * 16 + r + 8 * hi) * N + col0 + m] = v;
    }
}

// Dual-K strip GEMM: C = A1*B1 + A2*B2 + bias (LSTM pre-activation).
template<int K1, int K2, int N>
__global__ void strip_gemm_dual_kernel(const float* __restrict__ A1,
                                       const float* __restrict__ B1,
                                       const float* __restrict__ A2,
                                       const float* __restrict__ B2,
                                       const float* __restrict__ bias,
                                       float* __restrict__ C, int numWaves) {
  const int wid = blockIdx.x * (blockDim.x >> 5) + (threadIdx.x >> 5);
  if (wid >= numWaves) return;
  constexpr int NT = N / 16;
  const int sid = wid / NT, nt = wid % NT;
  const int row0 = sid * 128, col0 = nt * 16;
  const int lane = threadIdx.x & 31;
  const int m = lane & 15, hi = lane >> 4;

  v8f acc[8];
#pragma unroll
  for (int i = 0; i < 8; ++i) acc[i] = (v8f){};

  for (int k0 = 0; k0 < K1; k0 += 4) {
    v2f bf;
    bf.x = B1[(size_t)(k0 + hi) * N + col0 + m];
    bf.y = B1[(size_t)(k0 + 2 + hi) * N + col0 + m];
#pragma unroll
    for (int i = 0; i < 8; ++i) {
      v2f af = *(const v2f*)(A1 + (size_t)(row0 + i * 16 + m) * K1 + k0 + 2 * hi);
      acc[i] = wmma4(af, bf, acc[i]);
    }
  }
  for (int k0 = 0; k0 < K2; k0 += 4) {
    v2f bf;
    bf.x = B2[(size_t)(k0 + hi) * N + col0 + m];
    bf.y = B2[(size_t)(k0 + 2 + hi) * N + col0 + m];
#pragma unroll
    for (int i = 0; i < 8; ++i) {
      v2f af = *(const v2f*)(A2 + (size_t)(row0 + i * 16 + m) * K2 + k0 + 2 * hi);
      acc[i] = wmma4(af, bf, acc[i]);
    }
  }
  const float bv = bias[col0 + m];
#pragma unroll
  for (int i = 0; i < 8; ++i)
#pragma unroll
    for (int r = 0; r < 8; ++r)
      C[(size_t)(row0 + i * 16 + r + 8 * hi) * N + col0 + m] = acc[i][r] + bv;
}

// ---------------------------------------------------------------------------
// Vocab logits GEMM: C[128,20000] = yp[128,128] @ Wo[128,20000] + bo.
// The A matrix (yp, 64KB fp32) is staged into LDS once per block by the
// Tensor Data Mover (tensor_load_to_lds, 2D descriptor per ISA ch.8), then
// A-fragments come from ds_load_b64 while B streams from L2.
// 125 blocks x 10 waves covers the 1250 N-tiles exactly (no guard needed,
// so the block barrier is safe).
// ---------------------------------------------------------------------------
__global__ __launch_bounds__(320)
void logits_gemm_kernel(const float* __restrict__ yp,
                        const float* __restrict__ Wo,
                        const float* __restrict__ bo,
                        float* __restrict__ C) {
  __shared__ float sA[P_ * P_];                      // 64 KB
  const int wave = threadIdx.x >> 5;
  const int lane = threadIdx.x & 31;

  if (wave == 0) {
    union { unsigned long long u; unsigned w[2]; } ga;
    ga.u = (unsigned long long)(uintptr_t)yp;
    unsigned lds_off = (unsigned)(uintptr_t)(&sA[0]); // LDS aperture: low 32 bits
    v4u g0;
    g0.x = 1u;                                                    // count=1
    g0.y = (unsigned)__builtin_amdgcn_readfirstlane((int)lds_off);
    g0.z = (unsigned)__builtin_amdgcn_readfirstlane((int)ga.w[0]);
    g0.w = (unsigned)__builtin_amdgcn_readfirstlane(
               (int)((ga.w[1] & 0x01ffffffu) | 0x80000000u));     // type=2
    v8i g1;
    g1[0] = (int)(2u << 16);        // data_size = 2 -> 4-byte elements
    g1[1] = (int)(128u << 16);      // tensor_dim0 = 128 (bits 63:48)
    g1[2] = (int)(128u << 16);      // tensor_dim1 = 128 (bits 95:80)
    g1[3] = (int)(128u << 16);      // tile_dim0   = 128 (bits 127:112)
    g1[4] = 128;                    // tile_dim1   = 128 (bits 143:128)
    g1[5] = 128;                    // tensor_dim0_stride = 128 (bits 191:160)
    g1[6] = 0;
    g1[7] = 0;
    v4i g2 = {};
    v4i g3 = {};
    v8i g4 = {};                    // groups 2/3 disabled (2D tensor)
    __builtin_amdgcn_tensor_load_to_lds(g0, g1, g2, g3, g4, 0);
    __builtin_amdgcn_s_wait_tensorcnt(0);
  }
  __syncthreads();

  const int nt   = blockIdx.x * 10 + wave;           // 0..1249
  const int col0 = nt * 16;
  const int m = lane & 15, hi = lane >> 4;

  v8f acc[8];
#pragma unroll
  for (int i = 0; i < 8; ++i) acc[i] = (v8f){};

  for (int k0 = 0; k0 < P_; k0 += 4) {
    v2f bf;
    bf.x = Wo[(size_t)(k0 + hi) * VOC_ + col0 + m];
    bf.y = Wo[(size_t)(k0 + 2 + hi) * VOC_ + col0 + m];
#pragma unroll
    for (int i = 0; i < 8; ++i) {
      v2f af = *(const v2f*)(&sA[(i * 16 + m) * P_ + k0 + 2 * hi]);
      acc[i] = wmma4(af, bf, acc[i]);
    }
  }
  const float bv = bo[col0 + m];
#pragma unroll
  for (int i = 0; i < 8; ++i)
#pragma unroll
    for (int r = 0; r < 8; ++r)
      C[(size_t)(i * 16 + r + 8 * hi) * VOC_ + col0 + m] = acc[i][r] + bv;
}

// ---------------------------------------------------------------------------
// Attention + softmax(L) + context + concat(emb) + pre-LSTM LayerNorm
// ---------------------------------------------------------------------------
__device__ __forceinline__ float sigmoidf_(float x) { return 1.0f / (1.0f + expf(-x)); }

__global__ void attention_kernel(const float* __restrict__ fW1,
                                 const float* __restrict__ hW2,
                                 const float* __restrict__ features,
                                 const float* __restrict__ Vw,
                                 const float* __restrict__ Vb,
                                 const int*   __restrict__ caption,
                                 const float* __restrict__ Emb,
                                 const float* __restrict__ g_pre,
                                 const float* __restrict__ b_pre,
                                 float* __restrict__ x, int t) {
  const int b = blockIdx.x, tid = threadIdx.x;
  __shared__ float sh_h[A_];
  __shared__ float sw[L_];
  __shared__ float sctx[C_];
  __shared__ float sx[CE_];
  __shared__ float red[128], red2[128];

  for (int j = tid; j < A_; j += 128) sh_h[j] = hW2[(size_t)b * A_ + j];
  __syncthreads();

  if (tid < L_) {
    const float* f1 = fW1 + ((size_t)b * L_ + tid) * A_;
    float acc = 0.0f;
    for (int j = 0; j < A_; ++j) acc += tanhf(f1[j] + sh_h[j]) * Vw[j];
    sw[tid] = acc + Vb[0];
  }
  __syncthreads();

  if (tid == 0) {
    float mx = -1e30f;
    for (int l = 0; l < L_; ++l) mx = fmaxf(mx, sw[l]);
    float s = 0.0f;
    for (int l = 0; l < L_; ++l) { sw[l] = expf(sw[l] - mx); s += sw[l]; }
    float inv = 1.0f / s;
    for (int l = 0; l < L_; ++l) sw[l] *= inv;
  }
  __syncthreads();

  if (tid < C_) {
    float acc = 0.0f;
    const float* fb = features + (size_t)b * L_ * C_ + tid;
    for (int l = 0; l < L_; ++l) acc += sw[l] * fb[(size_t)l * C_];
    sctx[tid] = acc;
  }
  __syncthreads();

  const int token = caption[b * T_ + t];
  float s = 0.0f, s2 = 0.0f;
  for (int j = tid; j < CE_; j += 128) {
    float v = (j < C_) ? sctx[j] : Emb[(size_t)token * E_ + (j - C_)];
    sx[j] = v; s += v; s2 += v * v;
  }
  red[tid] = s; red2[tid] = s2; __syncthreads();
  for (int st = 64; st > 0; st >>= 1) {
    if (tid < st) { red[tid] += red[tid + st]; red2[tid] += red2[tid + st]; }
    __syncthreads();
  }
  const float mean = red[0] / (float)CE_;
  const float var  = red2[0] / (float)CE_ - mean * mean;
  const float rstd = rsqrtf(var + LN_EPS);
  for (int j = tid; j < CE_; j += 128)
    x[(size_t)b * CE_ + j] = (sx[j] - mean) * rstd * g_pre[j] + b_pre[j];
}

// ---------------------------------------------------------------------------
// LSTM gates (Keras i,f,g,o) + post-LSTM LayerNorm
// ---------------------------------------------------------------------------
__global__ void lstm_ln_kernel(const float* __restrict__ z,
                               const float* __restrict__ c_prev,
                               float* __restrict__ c_next,
                               float* __restrict__ h_next,
                               float* __restrict__ y,
                               const float* __restrict__ g_post,
                               const float* __restrict__ b_post) {
  const int b = blockIdx.x, tid = threadIdx.x;   // 256 threads
  __shared__ float sh[R_];
  __shared__ float red[256], red2[256];
  float s = 0.0f, s2 = 0.0f;
  const float* zb = z + (size_t)b * 4 * R_;
  for (int r = tid; r < R_; r += 256) {
    float zi = zb[r], zf = zb[R_ + r], zg = zb[2 * R_ + r], zo = zb[3 * R_ + r];
    float c  = c_prev[(size_t)b * R_ + r];
    float cn = sigmoidf_(zf) * c + sigmoidf_(zi) * tanhf(zg);
    float hn = sigmoidf_(zo) * tanhf(cn);
    c_next[(size_t)b * R_ + r] = cn;
    h_next[(size_t)b * R_ + r] = hn;
    sh[r] = hn; s += hn; s2 += hn * hn;
  }
  red[tid] = s; red2[tid] = s2; __syncthreads();
  for (int st = 128; st > 0; st >>= 1) {
    if (tid < st) { red[tid] += red[tid + st]; red2[tid] += red2[tid + st]; }
    __syncthreads();
  }
  const float mean = red[0] / (float)R_;
  const float var  = red2[0] / (float)R_ - mean * mean;
  const float rstd = rsqrtf(var + LN_EPS);
  for (int r = tid; r < R_; r += 256)
    y[(size_t)b * R_ + r] = (sh[r] - mean) * rstd * g_post[r] + b_post[r];
}

// ---------------------------------------------------------------------------
// Row softmax over VOC -> d_out[b, t, :]
// ---------------------------------------------------------------------------
__global__ void vocab_softmax_kernel(const float* __restrict__ logits,
                                     float* __restrict__ out, int t) {
  const int b = blockIdx.x, tid = threadIdx.x;
  __shared__ float red[256];
  const float* zr = logits + (size_t)b * VOC_;
  float mx = -1e30f;
  for (int j = tid; j < VOC_; j += 256) mx = fmaxf(mx, zr[j]);
  red[tid] = mx; __syncthreads();
  for (int st = 128; st > 0; st >>= 1) {
    if (tid < st) red[tid] = fmaxf(red[tid], red[tid + st]);
    __syncthreads();
  }
  mx = red[0]; __syncthreads();
  float s = 0.0f;
  for (int j = tid; j < VOC_; j += 256) s += expf(zr[j] - mx);
  red[tid] = s; __syncthreads();
  for (int st = 128; st > 0; st >>= 1) {
    if (tid < st) red[tid] += red[tid + st];
    __syncthreads();
  }
  const float inv = 1.0f / red[0];
  float* dst = out + ((size_t)b * T_ + t) * VOC_;
  for (int j = tid; j < VOC_; j += 256) dst[j] = expf(zr[j] - mx) * inv;
}

__global__ void zero_kernel(float* __restrict__ p, int n) {
  int i = blockIdx.x * blockDim.x + threadIdx.x;
  if (i < n) p[i] = 0.0f;
}

// ---------------------------------------------------------------------------
extern "C" void kernel_launch(void* const* d_in, const int* in_sizes, int n_in,
                              void* d_out, int out_size, void* d_ws, size_t ws_size,
                              hipStream_t stream) {
  const float* features = (const float*)d_in[0];
  const int*   caption  = (const int*)  d_in[1];
  const float* Emb      = (const float*)d_in[2];
  const float* W1       = (const float*)d_in[3];
  const float* b1       = (const float*)d_in[4];
  const float* W2       = (const float*)d_in[5];
  const float* b2       = (const float*)d_in[6];
  const float* Vw       = (const float*)d_in[7];
  const float* Vb       = (const float*)d_in[8];
  const float* ln_pre_g = (const float*)d_in[9];
  const float* ln_pre_b = (const float*)d_in[10];
  const float* Wk       = (const float*)d_in[11];
  const float* Uk       = (const float*)d_in[12];
  const float* bk       = (const float*)d_in[13];
  const float* ln_post_g= (const float*)d_in[14];
  const float* ln_post_b= (const float*)d_in[15];
  const float* Wp       = (const float*)d_in[16];
  const float* bp       = (const float*)d_in[17];
  const float* Wo       = (const float*)d_in[18];
  const float* bo       = (const float*)d_in[19];
  float* out = (float*)d_out;

  float* ws   = (float*)d_ws;
  float* fW1  = ws;                         // B*L*A
  float* h0   = fW1 + (size_t)B_ * L_ * A_;
  float* c0   = h0  + B_ * R_;
  float* h1   = c0  + B_ * R_;
  float* c1   = h1  + B_ * R_;
  float* hW2  = c1  + B_ * R_;
  float* xbuf = hW2 + B_ * A_;
  float* zbuf = xbuf + B_ * CE_;
  float* ybuf = zbuf + B_ * 4 * R_;
  float* ypbuf= ybuf + B_ * R_;
  float* zob  = ypbuf + B_ * P_;

  {
    int n = 2 * B_ * R_;                   // zero h0,c0 (contiguous)
    zero_kernel<<<(n + 255) / 256, 256, 0, stream>>>(h0, n);
  }

  // fW1 = features @ W1 + b1  : [(B*L)=8192 x 100] * [100 x 512], 64 strips
  {
    int waves = 64 * (A_ / 16);            // 2048
    strip_gemm_kernel<C_, A_, 0><<<(waves + 3) / 4, 128, 0, stream>>>(
        features, W1, b1, fW1, waves);
  }

  for (int t = 0; t < T_; ++t) {
    float* hp = (t & 1) ? h1 : h0;
    float* cp = (t & 1) ? c1 : c0;
    float* hn = (t & 1) ? h0 : h1;
    float* cn = (t & 1) ? c0 : c1;

    // hW2 = h @ W2 + b2 : [128x512]*[512x512], 32 waves
    strip_gemm_kernel<R_, A_, 0><<<8, 128, 0, stream>>>(hp, W2, b2, hW2, 32);

    attention_kernel<<<B_, 128, 0, stream>>>(fW1, hW2, features, Vw, Vb,
                                             caption, Emb, ln_pre_g, ln_pre_b,
                                             xbuf, t);

    // z = x@Wk + h@Uk + bk : [128 x 2048], 128 waves
    strip_gemm_dual_kernel<CE_, R_, 4 * R_><<<32, 128, 0, stream>>>(
        xbuf, Wk, hp, Uk, bk, zbuf, 128);

    lstm_ln_kernel<<<B_, 256, 0, stream>>>(zbuf, cp, cn, hn, ybuf,
                                           ln_post_g, ln_post_b);

    // yp = relu(y @ Wp + bp) : [128x512]*[512x128], 8 waves
    strip_gemm_kernel<R_, P_, 1><<<2, 128, 0, stream>>>(ybuf, Wp, bp, ypbuf, 8);

    // logits = yp @ Wo + bo : TDM-staged A in LDS, 125 blocks x 10 waves
    logits_gemm_kernel<<<125, 320, 0, stream>>>(ypbuf, Wo, bo, zob);

    vocab_softmax_kernel<<<B_, 256, 0, stream>>>(zob, out, t);
  }
}